// MessageBuildingLayerLSH_33724083208429
// MI455X (gfx1250) — compile-verified
//
#include <hip/hip_runtime.h>
#include <hip/hip_bf16.h>
#include <stdint.h>

// Problem constants (from reference setup_inputs)
#define BATCH   16
#define NPTS    16384
#define DMSG    128
#define DNODE   256
#define NBINS   128      // NPTS / BIN_SIZE
#define BINSZ   128
#define NCOLS   64       // n_bins // 2
#define CB_STRIDE 100    // codebook is [128, 100], we use first 64 columns

typedef __attribute__((ext_vector_type(2))) float v2f;
typedef __attribute__((ext_vector_type(8))) float v8f;

#if defined(__has_builtin)
#if __has_builtin(__builtin_amdgcn_wmma_f32_16x16x4_f32)
#define HAVE_WMMA_F32X4 1
#endif
#endif
#ifndef HAVE_WMMA_F32X4
#define HAVE_WMMA_F32X4 0
#endif

// ---------------------------------------------------------------------------
// Kernel 1: LSH codes + argmax -> bin_idx[b][n]
// One block of 64 threads handles 64 points; x_msg rows staged in LDS (32 KB).
// ---------------------------------------------------------------------------
__global__ void __launch_bounds__(64)
lsh_bin_kernel(const float* __restrict__ x_msg,
               const uint8_t* __restrict__ msk,
               const float* __restrict__ cb,
               int* __restrict__ bin_idx)
{
    __shared__ float Xs[64 * DMSG];  // 32 KB

    const int groups_per_batch = NPTS / 64;
    const int b  = blockIdx.x / groups_per_batch;
    const int g  = blockIdx.x % groups_per_batch;
    const int p0 = g * 64;

    const float* src = x_msg + ((size_t)b * NPTS + p0) * DMSG;
    for (int e = threadIdx.x; e < 64 * DMSG; e += 64)
        Xs[e] = src[e];  // coalesced: consecutive threads -> consecutive addrs
    __syncthreads();

    const int t = threadIdx.x;
    const float* xr = &Xs[t * DMSG];

    // argmax over [mul, -mul]; ties resolve to first occurrence (i.e. +mul side)
    float bp = -3.402823466e+38f; int ap = 0;  // best over +mul
    float bn = -3.402823466e+38f; int an = 0;  // best over -mul
    #pragma unroll 1
    for (int c = 0; c < NCOLS; ++c) {
        float acc = 0.0f;
        #pragma unroll 8
        for (int k = 0; k < DMSG; ++k)
            acc += xr[k] * cb[k * CB_STRIDE + c];   // same addr across lanes -> broadcast
        if ( acc > bp) { bp =  acc; ap = c; }
        if (-acc > bn) { bn = -acc; an = c; }
    }
    int arg = (bp >= bn) ? ap : (an + NCOLS);

    const int p = p0 + t;
    const int penalty = msk[(size_t)b * NPTS + p] ? 0 : (NBINS - 1);
    bin_idx[(size_t)b * NPTS + p] = arg + penalty;
}

// ---------------------------------------------------------------------------
// Kernel 2: stable counting sort of bin_idx per batch -> sorted indices.
// One block (256 threads) per batch. Values are in [0, 254].
// ---------------------------------------------------------------------------
__global__ void __launch_bounds__(256)
bin_sort_kernel(const int* __restrict__ bin_idx,
                int* __restrict__ sorted,
                float* __restrict__ out_bins)
{
    __shared__ int counts[256];
    __shared__ int offs[256];
    __shared__ int chunk[2048];

    const int b = blockIdx.x;
    const int t = threadIdx.x;
    const int* bi = bin_idx + (size_t)b * NPTS;

    counts[t] = 0;
    __syncthreads();
    for (int i = t; i < NPTS; i += 256)
        atomicAdd(&counts[bi[i]], 1);
    __syncthreads();
    if (t == 0) {
        int run = 0;
        for (int v = 0; v < 256; ++v) { offs[v] = run; run += counts[v]; }
    }
    __syncthreads();

    int cursor = offs[t];  // thread t owns bin value t (t==255 is empty)
    for (int c = 0; c < NPTS / 2048; ++c) {
        for (int j = 0; j < 2048; j += 256)
            chunk[j + t] = bi[c * 2048 + j + t];
        __syncthreads();
        if (t < 255) {
            for (int i = 0; i < 2048; ++i) {
                if (chunk[i] == t) {
                    const int pos = cursor++;
                    const int src = c * 2048 + i;
                    sorted[(size_t)b * NPTS + pos]   = src;
                    out_bins[(size_t)b * NPTS + pos] = (float)src;
                }
            }
        }
        __syncthreads();
    }
}

// ---------------------------------------------------------------------------
// Kernel 3: per (batch, bin): gather masked x_msg tile to LDS, Gram matrix via
// V_WMMA_F32_16X16X4_F32, distance->exp->mask, plus x_node gather + msk_b out.
// 256 threads = 8 waves; 64 WMMA output tiles of 16x16, 8 per wave.
// ---------------------------------------------------------------------------
#define XS 132  // padded LDS row stride (floats) to avoid bank conflicts

__global__ void __launch_bounds__(256)
bin_gather_gram_kernel(const float* __restrict__ x_msg,
                       const float* __restrict__ x_node,
                       const uint8_t* __restrict__ msk,
                       const int* __restrict__ sorted,
                       float* __restrict__ out_feat,
                       float* __restrict__ out_dm,
                       float* __restrict__ out_msk)
{
    extern __shared__ char smem_raw[];
    float* Xsh = (float*)smem_raw;          // 128 * XS floats (masked x_msg tile)
    float* nrm = Xsh + BINSZ * XS;          // 128 row norms
    float* mvs = nrm + BINSZ;               // 128 mask values
    int*   idx = (int*)(mvs + BINSZ);       // 128 gathered indices

    const int bidx = blockIdx.x;
    const int b    = bidx / NBINS;
    const int bin  = bidx % NBINS;
    const int t    = threadIdx.x;

    // gathered indices + mask (+ msk_b output)
    if (t < BINSZ) {
        const int j = sorted[((size_t)b * NBINS + bin) * BINSZ + t];
        idx[t] = j;
        const float m = msk[(size_t)b * NPTS + j] ? 1.0f : 0.0f;
        mvs[t] = m;
        out_msk[((size_t)b * NBINS + bin) * BINSZ + t] = m;
    }
    __syncthreads();

    // gather masked x_msg rows into LDS
    for (int e = t; e < BINSZ * DMSG; e += 256) {
        const int r = e >> 7, c = e & 127;
        Xsh[r * XS + c] = x_msg[((size_t)b * NPTS + idx[r]) * DMSG + c] * mvs[r];
    }
    __syncthreads();

    // row norms of masked tile
    if (t < BINSZ) {
        float acc = 0.0f;
        #pragma unroll 8
        for (int k = 0; k < DMSG; ++k) { const float x = Xsh[t * XS + k]; acc += x * x; }
        nrm[t] = acc;
    }
    __syncthreads();

    const int wave  = t >> 5;
    const int lane  = t & 31;
    const int lm    = lane & 15;          // A: row M, B: col N
    const int khalf = (lane >> 4) << 1;   // lanes 16-31 hold K+2 (ISA 16x4 f32 layout)
    const int mhi   = (lane >> 4) << 3;   // C: lanes 16-31 hold M+8

    float* dmb = out_dm + ((size_t)b * NBINS + bin) * BINSZ * BINSZ;

    for (int tile = wave; tile < 64; tile += 8) {
        const int ti = tile >> 3, tj = tile & 7;
        const float* arow = Xsh + (ti * 16 + lm) * XS;  // A[m=lm][k]
        const float* brow = Xsh + (tj * 16 + lm) * XS;  // B[k][n=lm] = X[n][k]
        v8f c = {};
#if HAVE_WMMA_F32X4
        #pragma unroll 4
        for (int k0 = 0; k0 < DMSG; k0 += 4) {
            v2f a, bb;
            a.x  = arow[k0 + khalf];
            a.y  = arow[k0 + khalf + 1];
            bb.x = brow[k0 + khalf];
            bb.y = brow[k0 + khalf + 1];
            // D = A(16x4) * B(4x16) + C, full f32
            c = __builtin_amdgcn_wmma_f32_16x16x4_f32(
                    false, a, false, bb, (short)0, c, false, false);
        }
#else
        // scalar fallback (same results, no WMMA)
        #pragma unroll
        for (int r = 0; r < 8; ++r) {
            const int ig = ti * 16 + r + mhi;
            const int jg = tj * 16 + lm;
            float acc = 0.0f;
            for (int k = 0; k < DMSG; ++k)
                acc += Xsh[ig * XS + k] * Xsh[jg * XS + k];
            c[r] = acc;
        }
#endif
        const int   jg = tj * 16 + lm;
        const float nj = nrm[jg];
        const float mj = mvs[jg];
        #pragma unroll
        for (int r = 0; r < 8; ++r) {
            const int ig = ti * 16 + r + mhi;
            float d2 = nrm[ig] - 2.0f * c[r] + nj;
            d2 = fminf(fmaxf(d2, 1e-6f), 1e6f);
            float w = expf(-0.1f * sqrtf(d2));
            w = fminf(fmaxf(w, 0.0f), 1.0f);
            w *= mvs[ig] * mj;
            dmb[(size_t)ig * BINSZ + jg] = w;
        }
    }

    // gather x_node rows (float4 vectorized, coalesced)
    for (int e = t; e < BINSZ * (DNODE / 4); e += 256) {
        const int r  = e >> 6;
        const int c4 = e & 63;
        const float4 v =
            ((const float4*)(x_node + ((size_t)b * NPTS + idx[r]) * DNODE))[c4];
        ((float4*)(out_feat + ((size_t)b * NBINS + bin) * BINSZ * DNODE
                            + (size_t)r * DNODE))[c4] = v;
    }
}

// ---------------------------------------------------------------------------
// Launch: inputs = {x_msg f32, x_node f32, msk bool, codebook f32}
// Outputs (flat floats, concatenated): bins_split | x_feat_b | dm | msk_b
// ---------------------------------------------------------------------------
extern "C" void kernel_launch(void* const* d_in, const int* in_sizes, int n_in,
                              void* d_out, int out_size, void* d_ws, size_t ws_size,
                              hipStream_t stream) {
    const float*   x_msg  = (const float*)d_in[0];
    const float*   x_node = (const float*)d_in[1];
    const uint8_t* msk    = (const uint8_t*)d_in[2];
    const float*   cb     = (const float*)d_in[3];

    float* out = (float*)d_out;
    const size_t n_bins_out = (size_t)BATCH * NBINS * BINSZ;           // 262,144
    const size_t n_feat_out = (size_t)BATCH * NBINS * BINSZ * DNODE;   // 67,108,864
    const size_t n_dm_out   = (size_t)BATCH * NBINS * BINSZ * BINSZ;   // 33,554,432
    float* out_bins = out;
    float* out_feat = out + n_bins_out;
    float* out_dm   = out_feat + n_feat_out;
    float* out_msk  = out_dm + n_dm_out;

    int* bin_idx = (int*)d_ws;              // BATCH*NPTS ints (1 MB)
    int* sorted  = bin_idx + BATCH * NPTS;  // BATCH*NPTS ints (1 MB)

    lsh_bin_kernel<<<BATCH * (NPTS / 64), 64, 0, stream>>>(x_msg, msk, cb, bin_idx);

    bin_sort_kernel<<<BATCH, 256, 0, stream>>>(bin_idx, sorted, out_bins);

    const size_t smem = (size_t)(BINSZ * XS + BINSZ + BINSZ) * sizeof(float)
                      + BINSZ * sizeof(int);   // ~67.5 KB (69,120 B) of the 320 KB WGP LDS
    bin_gather_gram_kernel<<<BATCH * NBINS, 256, smem, stream>>>(
        x_msg, x_node, msk, sorted, out_feat, out_dm, out_msk);
}